// ChamferCUDA_61194694033711
// MI455X (gfx1250) — compile-verified
//
#include <hip/hip_runtime.h>

typedef float v2f __attribute__((ext_vector_type(2)));
typedef float v8f __attribute__((ext_vector_type(8)));

#define B_ 16
#define N_ 4096
#define M_ 4096
#define AVG_SCALE 0.5f
#define BIGF 3.0e38f

// ---------------------------------------------------------------------------
// Kernel 1: initialize workspace (dist2 partial-min array = +huge, s1 = 0)
// ---------------------------------------------------------------------------
__global__ void chamfer_init(float* __restrict__ s1_total,
                             float* __restrict__ dist2) {
    int i = blockIdx.x * blockDim.x + threadIdx.x;
    if (i == 0) *s1_total = 0.0f;
    if (i < B_ * M_) dist2[i] = BIGF;
}

// ---------------------------------------------------------------------------
// Kernel 2: WMMA chamfer. One wave32 per (batch, 32-row block of points1).
// Two 16-row A-tiles share every B-tile; two B-tiles in flight per loop
// iteration (4 WMMAs / 2 loads / 2 atomics), software-pipelined prefetch.
//   A (16x4):  row n -> [-2x, -2y, -2z, 1]
//   B (4x16):  col m -> [x', y', z', ||q||^2]
//   D = A*B   = -2 p.q + ||q_m||^2  per 16x16 tile.
//   t = D + rownorm  done as one v8f add (4 aligned v_pk_add_f32, no
//   register-pairing movs); t feeds both dist1 row-min and dist2 col-min.
// ISA layouts (CDNA5 ISA §7.12.2):
//   A/B: lanes 0-15 hold K=0 (v0) / K=1 (v1); lanes 16-31 hold K=2 / K=3.
//   D:   VGPR v, lane L -> row (v + 8*(L>=16)), col (L & 15).
// ---------------------------------------------------------------------------
__global__ __launch_bounds__(32)
void chamfer_wmma(const float* __restrict__ p1,
                  const float* __restrict__ p2,
                  float* __restrict__ s1_total,
                  float* __restrict__ dist2) {
    const int lane = threadIdx.x;          // 0..31
    const int b    = blockIdx.x >> 7;      // 128 32-row blocks per batch
    const int nb   = blockIdx.x & 127;
    const int n0   = nb << 5;              // first of 32 rows
    const int r    = lane & 15;
    const int hi   = lane >> 4;            // 0 = lanes 0-15, 1 = lanes 16-31

    // ---- load my two points1 rows (halves load the same 16 points) ----
    const float* p1pA = p1 + ((size_t)b * N_ + n0 + r) * 3;
    const float* p1pB = p1pA + 16 * 3;
    const float xA = p1pA[0], yA = p1pA[1], zA = p1pA[2];
    const float xB = p1pB[0], yB = p1pB[1], zB = p1pB[2];
    const float sqA = xA * xA + yA * yA + zA * zA;
    const float sqB = xB * xB + yB * yB + zB * zB;

    // A-matrix fragments (2 VGPRs each)
    v2f a0, a1;
    a0.x = hi ? (-2.0f * zA) : (-2.0f * xA);   // K=2 : K=0
    a0.y = hi ? 1.0f         : (-2.0f * yA);   // K=3 : K=1
    a1.x = hi ? (-2.0f * zB) : (-2.0f * xB);
    a1.y = hi ? 1.0f         : (-2.0f * yB);

    // per-lane ||p||^2 for the 8 rows each D-fragment covers, as v8f so the
    // "+ rownorm" is a tuple-aligned packed add against the WMMA dest tuple
    v8f sqRowA, sqRowB;
#pragma unroll
    for (int v = 0; v < 8; ++v) {
        sqRowA[v] = __shfl(sqA, v + 8 * hi, 32);
        sqRowB[v] = __shfl(sqB, v + 8 * hi, 32);
    }

    const float* p2b = p2 + (size_t)b * M_ * 3;
    float* d2base = dist2 + (size_t)b * M_ + r;

    v8f rminA, rminB;    // running min over m of (D + rownorm) == dist1/row
#pragma unroll
    for (int v = 0; v < 8; ++v) { rminA[v] = BIGF; rminB[v] = BIGF; }

    // one B-tile: 2 WMMAs + min network + dist2 atomic (all 32 lanes;
    // lanes L and L+16 share an address, hardware min resolves it)
    auto tile = [&](int m0, float px, float py, float pz) {
        const float sq2 = px * px + py * py + pz * pz;
        v2f bb;
        bb.x = hi ? pz  : px;    // K=2 : K=0
        bb.y = hi ? sq2 : py;    // K=3 : K=1

        v8f c0 = {}, c1 = {};
        c0 = __builtin_amdgcn_wmma_f32_16x16x4_f32(
                false, a0, false, bb, (short)0, c0, false, false);
        c1 = __builtin_amdgcn_wmma_f32_16x16x4_f32(
                false, a1, false, bb, (short)0, c1, false, false);

        // full d2 values (row norm folded): one 8-wide add per fragment
        const v8f tA = c0 + sqRowA;     // rows v+8hi
        const v8f tB = c1 + sqRowB;     // rows 16+v+8hi

        float colmin;
#pragma unroll
        for (int v = 0; v < 8; ++v) {
            rminA[v] = fminf(rminA[v], tA[v]);
            rminB[v] = fminf(rminB[v], tB[v]);
            const float tm = fminf(tA[v], tB[v]);
            colmin = (v == 0) ? tm : fminf(colmin, tm);
        }

        float* addr = d2base + m0;
        asm volatile("global_atomic_min_num_f32 %0, %1, off scope:SCOPE_DEV"
                     :: "v"(addr), "v"(colmin)
                     : "memory");
    };

    // ---- software pipeline: 2 tiles in flight, incrementing prefetch ptr ----
    const float* q = p2b + (size_t)r * 3;
    float c0x = q[0],      c0y = q[1],      c0z = q[2];       // tile 0
    float c1x = q[48 + 0], c1y = q[48 + 1], c1z = q[48 + 2];  // tile 1
    const float* qp = q + 96;                                 // tile 2

    for (int m0 = 0; m0 <= M_ - 64; m0 += 32) {
        const float n0x = qp[0],      n0y = qp[1],      n0z = qp[2];
        const float n1x = qp[48 + 0], n1y = qp[48 + 1], n1z = qp[48 + 2];
        qp += 96;

        tile(m0,      c0x, c0y, c0z);
        tile(m0 + 16, c1x, c1y, c1z);

        c0x = n0x; c0y = n0y; c0z = n0z;
        c1x = n1x; c1y = n1y; c1z = n1z;
    }
    tile(M_ - 32, c0x, c0y, c0z);
    tile(M_ - 16, c1x, c1y, c1z);

    // ---- finish dist1: min across the 16 lanes of each half ----
    float s = 0.0f;
#pragma unroll
    for (int v = 0; v < 8; ++v) {
        float x = rminA[v];
        x = fminf(x, __shfl_xor(x, 1, 32));
        x = fminf(x, __shfl_xor(x, 2, 32));
        x = fminf(x, __shfl_xor(x, 4, 32));
        x = fminf(x, __shfl_xor(x, 8, 32));
        s += x;                              // dist1[row v + 8*hi]

        float y = rminB[v];
        y = fminf(y, __shfl_xor(y, 1, 32));
        y = fminf(y, __shfl_xor(y, 2, 32));
        y = fminf(y, __shfl_xor(y, 4, 32));
        y = fminf(y, __shfl_xor(y, 8, 32));
        s += y;                              // dist1[row 16 + v + 8*hi]
    }
    // lower half lanes: rows {0-7, 16-23}; upper half: rows {8-15, 24-31}
    s += __shfl_xor(s, 16, 32);
    if (lane == 0) atomicAdd(s1_total, s);
}

// ---------------------------------------------------------------------------
// Kernel 3: final reduction -> scalar output
//   out = 0.5 * ( sum(dist1)/N + sum(dist2)/M )
// ---------------------------------------------------------------------------
__global__ __launch_bounds__(256)
void chamfer_reduce(const float* __restrict__ s1_total,
                    const float* __restrict__ dist2,
                    float* __restrict__ out) {
    __shared__ float sm[256];
    float acc = 0.0f;
    for (int i = threadIdx.x; i < B_ * M_; i += 256) acc += dist2[i];
    sm[threadIdx.x] = acc;
    __syncthreads();
    for (int s = 128; s > 0; s >>= 1) {
        if (threadIdx.x < s) sm[threadIdx.x] += sm[threadIdx.x + s];
        __syncthreads();
    }
    if (threadIdx.x == 0) {
        out[0] = AVG_SCALE * (s1_total[0] / (float)N_ + sm[0] / (float)M_);
    }
}

// ---------------------------------------------------------------------------
extern "C" void kernel_launch(void* const* d_in, const int* in_sizes, int n_in,
                              void* d_out, int out_size, void* d_ws, size_t ws_size,
                              hipStream_t stream) {
    const float* p1 = (const float*)d_in[0];   // [B, N, 3] f32
    const float* p2 = (const float*)d_in[1];   // [B, M, 3] f32
    float* out = (float*)d_out;                // scalar f32

    // ws layout: [0..4) s1_total, dist2 array at +256 bytes (aligned)
    float* s1    = (float*)d_ws;
    float* dist2 = (float*)((char*)d_ws + 256);

    const int initN = B_ * M_;
    chamfer_init<<<(initN + 255) / 256, 256, 0, stream>>>(s1, dist2);
    chamfer_wmma<<<B_ * (N_ / 32), 32, 0, stream>>>(p1, p2, s1, dist2);
    chamfer_reduce<<<1, 256, 0, stream>>>(s1, dist2, out);
}